// GraphAttentionLayer_16595753631946
// MI455X (gfx1250) — compile-verified
//
#include <hip/hip_runtime.h>
#include <cmath>

#define IN_F   128
#define OUT_F  64
#define NNODE  2048
#define NB     8
#define LRALPHA 0.2f
#define NEG_INF -9000000000000000.0f

typedef __attribute__((ext_vector_type(2))) float v2f;
typedef __attribute__((ext_vector_type(8))) float v8f;

// D(16x16,f32) = A(16x4,f32) x B(4x16,f32) + C   -- CDNA5 V_WMMA_F32_16X16X4_F32
// Lane layout (ISA 7.12.2): m = lane&15, kh = lane>>4
//   A: a[v] = A[m][2*kh+v]      B: b[v] = B[2*kh+v][m]
//   D: d[r] = D[r + 8*kh][m]
__device__ __forceinline__ v8f wmma_f32_4(v2f a, v2f b, v8f c) {
  return __builtin_amdgcn_wmma_f32_16x16x4_f32(false, a, false, b, (short)0, c,
                                               false, false);
}

// ---------------------------------------------------------------- Kernel 1
// Wh[b,n,:] = h[b,n,:] @ W.  One wave -> 16 rows x 64 cols.
// W staged in LDS in paired-K layout: Wsh[p*128 + c*2 + v] = W[2p+v][c]
// so each WMMA B fragment is one aligned ds_load_b64.
__global__ __launch_bounds__(256) void k1_wh(const float* __restrict__ h,
                                             const float* __restrict__ W,
                                             float* __restrict__ Wh) {
  __shared__ __align__(16) float Wsh[(IN_F / 2) * 128];  // 32 KB, paired
  const int tid = threadIdx.x;
#pragma unroll
  for (int t = 0; t < 8; ++t) {
    const int flat = t * 256 + tid;        // (pair p, col-pair c2) items
    const int p = flat >> 5, c2 = flat & 31;
    const float2 r0 = *(const float2*)&W[(2 * p) * OUT_F + c2 * 2];
    const float2 r1 = *(const float2*)&W[(2 * p + 1) * OUT_F + c2 * 2];
    const float4 o = {r0.x, r1.x, r0.y, r1.y};
    *(float4*)&Wsh[p * 128 + c2 * 4] = o;
  }
  __syncthreads();

  const int wave = tid >> 5, lane = tid & 31;
  const int m = lane & 15, kh = lane >> 4;
  const int gi0 = (blockIdx.x * 8 + wave) * 16;
  const float* hrow = h + (size_t)(gi0 + m) * IN_F;

  v8f acc[4] = {};
  for (int k0 = 0; k0 < IN_F; k0 += 4) {
    const float2 av = *(const float2*)&hrow[k0 + 2 * kh];
    v2f a; a.x = av.x; a.y = av.y;
    const float* wb = &Wsh[((k0 >> 1) + kh) * 128];
#pragma unroll
    for (int nt = 0; nt < 4; ++nt) {
      const v2f bv = *(const v2f*)&wb[(nt * 16 + m) * 2];  // ds_load_b64
      acc[nt] = wmma_f32_4(a, bv, acc[nt]);
    }
  }
#pragma unroll
  for (int nt = 0; nt < 4; ++nt)
#pragma unroll
    for (int r = 0; r < 8; ++r)
      Wh[(size_t)(gi0 + r + 8 * kh) * OUT_F + nt * 16 + m] = acc[nt][r];
}

// ---------------------------------------------------------------- Kernel 2
// f1[row] = Wh[row,:] . a[:64] ; f2[row] = Wh[row,:] . a[64:128]
__global__ __launch_bounds__(256) void k2_f(const float* __restrict__ Wh,
                                            const float* __restrict__ avec,
                                            float* __restrict__ f1,
                                            float* __restrict__ f2) {
  const int tid = threadIdx.x, wave = tid >> 5, lane = tid & 31;
  const int row = blockIdx.x * 8 + wave;
  const float2 w  = *(const float2*)&Wh[(size_t)row * OUT_F + 2 * lane];
  const float2 a1 = *(const float2*)&avec[2 * lane];
  const float2 a2 = *(const float2*)&avec[OUT_F + 2 * lane];
  float s1 = w.x * a1.x + w.y * a1.y;
  float s2 = w.x * a2.x + w.y * a2.y;
#pragma unroll
  for (int off = 16; off > 0; off >>= 1) {
    s1 += __shfl_xor(s1, off, 32);
    s2 += __shfl_xor(s2, off, 32);
  }
  if (lane == 0) { f1[row] = s1; f2[row] = s2; }
}

// ---------------------------------------------------------------- Kernel 3
// Online softmax statistics per row (one wave/row):
// rmax, rsum over score_j = adj>0 ? lrelu(f1_i + f2_j) : NEG_INF
__global__ __launch_bounds__(256) void k3_stats(const int* __restrict__ adj,
                                                const float* __restrict__ f1,
                                                const float* __restrict__ f2,
                                                float* __restrict__ rmax,
                                                float* __restrict__ rsum) {
  const int tid = threadIdx.x, wave = tid >> 5, lane = tid & 31;
  const int row = blockIdx.x * 8 + wave;                  // 0..16383
  const float f1i = f1[row];
  const int* adjrow = adj + (size_t)row * NNODE;
  const float* f2b = f2 + (size_t)(row >> 11) * NNODE;    // batch base

  float mloc = NEG_INF, sloc = 0.0f;
  for (int t = 0; t < NNODE / 128; ++t) {
    const int jb = t * 128 + lane * 4;
    if (jb + 512 < NNODE)                                  // stream-ahead hint
      __builtin_prefetch(&adjrow[jb + 512], 0, 0);         // global_prefetch_b8
    const int4   av = *(const int4*)&adjrow[jb];
    const float4 fv = *(const float4*)&f2b[jb];
    float fs[4] = {fv.x, fv.y, fv.z, fv.w};
    int   as[4] = {av.x, av.y, av.z, av.w};
#pragma unroll
    for (int v = 0; v < 4; ++v) {
      float e = f1i + fs[v];
      e = fmaxf(e, LRALPHA * e);                           // leaky relu
      const float x = (as[v] > 0) ? e : NEG_INF;
      if (x > mloc) { sloc *= __expf(mloc - x); mloc = x; }
      sloc += __expf(x - mloc);
    }
  }
#pragma unroll
  for (int off = 16; off > 0; off >>= 1) {
    const float mo = __shfl_xor(mloc, off, 32);
    const float so = __shfl_xor(sloc, off, 32);
    const float M = fmaxf(mloc, mo);
    sloc = sloc * __expf(mloc - M) + so * __expf(mo - M);
    mloc = M;
  }
  if (lane == 0) { rmax[row] = mloc; rsum[row] = sloc; }
}

// ---------------------------------------------------------------- Kernel 4
// out[b,i,:] = elu( softmax(score) @ Wh ).  One wave -> 16 i-rows x 64 cols;
// 8 waves/block share one batch + lockstep j chunks. Per 64-wide chunk:
//   - adj[16][64] staged per wave (coalesced int4)
//   - Wh[j0:j0+64,:] staged ONCE per block in paired-K LDS layout (16 KB)
//   - f2 chunk staged once per block
// A fragment = exp(score - m_i) * 1/s_i built from two ds_load_b64;
// B fragment = one ds_load_b64; fp32 WMMA accumulates.
__global__ __launch_bounds__(256) void k4_out(const int* __restrict__ adj,
                                              const float* __restrict__ Wh,
                                              const float* __restrict__ f1,
                                              const float* __restrict__ f2,
                                              const float* __restrict__ rmax,
                                              const float* __restrict__ rsum,
                                              float* __restrict__ out) {
  __shared__ __align__(16) int   adjsh[8][16][64];  // 32 KB (per-wave tiles)
  __shared__ __align__(16) float whsh[32 * 128];    // 16 KB paired-K, shared
  __shared__ __align__(16) float f2sh[64];          // shared
  const int tid = threadIdx.x, wave = tid >> 5, lane = tid & 31;
  const int m = lane & 15, kh = lane >> 4;
  const int gi0 = (blockIdx.x * 8 + wave) * 16;     // flattened row base
  const int b = gi0 >> 11;
  const int myrow = gi0 + m;

  const float f1i  = f1[myrow];
  const float mi   = rmax[myrow];
  const float invs = 1.0f / rsum[myrow];
  const float* f2b = f2 + (size_t)b * NNODE;
  const float* Whb = Wh + (size_t)b * NNODE * OUT_F;

  v8f acc[4] = {};
  for (int j0 = 0; j0 < NNODE; j0 += 64) {
    __syncthreads();  // previous chunk fully consumed before restaging
    // ---- stage adj[16][64] for this wave's 16 rows
#pragma unroll
    for (int t = 0; t < 8; ++t) {
      const int flat = t * 32 + lane;               // int4 units 0..255
      const int r = flat >> 4, c = flat & 15;
      const size_t gofs = (size_t)(gi0 + r) * NNODE + j0 + c * 4;
      if (j0 + 64 < NNODE)
        __builtin_prefetch(&adj[gofs + 64], 0, 0);  // next chunk
      *(int4*)&adjsh[wave][r][c * 4] = *(const int4*)&adj[gofs];
    }
    // ---- stage Wh chunk (shared by all 8 waves), paired-K layout:
    //      whsh[p*128 + c*2 + v] = Wh[b, j0 + 2p + v, c]
#pragma unroll
    for (int t = 0; t < 4; ++t) {
      const int flat = t * 256 + tid;               // 0..1023 (p, c2) items
      const int p = flat >> 5, c2 = flat & 31;
      const float* src = Whb + (size_t)(j0 + 2 * p) * OUT_F + c2 * 2;
      const float2 r0 = *(const float2*)src;
      const float2 r1 = *(const float2*)(src + OUT_F);
      const float4 o = {r0.x, r1.x, r0.y, r1.y};
      *(float4*)&whsh[p * 128 + c2 * 4] = o;
    }
    if (tid < 16)
      *(float4*)&f2sh[tid * 4] = *(const float4*)&f2b[j0 + tid * 4];
    __syncthreads();

#pragma unroll 4
    for (int jj = 0; jj < 64; jj += 4) {
      // A fragment: two probabilities, inputs via two ds_load_b64
      const int2   ad  = *(const int2*)&adjsh[wave][m][jj + 2 * kh];
      const float2 f2v = *(const float2*)&f2sh[jj + 2 * kh];
      v2f a;
      {
        float e0 = f1i + f2v.x;           e0 = fmaxf(e0, LRALPHA * e0);
        float e1 = f1i + f2v.y;           e1 = fmaxf(e1, LRALPHA * e1);
        const float x0 = (ad.x > 0) ? e0 : NEG_INF;
        const float x1 = (ad.y > 0) ? e1 : NEG_INF;
        a.x = __expf(x0 - mi) * invs;
        a.y = __expf(x1 - mi) * invs;
      }
      const float* wb = &whsh[((jj >> 1) + kh) * 128];
#pragma unroll
      for (int nt = 0; nt < 4; ++nt) {
        const v2f bv = *(const v2f*)&wb[(nt * 16 + m) * 2];  // ds_load_b64
        acc[nt] = wmma_f32_4(a, bv, acc[nt]);
      }
    }
  }

#pragma unroll
  for (int nt = 0; nt < 4; ++nt)
#pragma unroll
    for (int r = 0; r < 8; ++r) {
      const float x = acc[nt][r];
      const float y = (x > 0.0f) ? x : expm1f(x);   // ELU (alpha=1)
      out[(size_t)(gi0 + r + 8 * kh) * OUT_F + nt * 16 + m] = y;
    }
}

// ---------------------------------------------------------------- launch
extern "C" void kernel_launch(void* const* d_in, const int* in_sizes, int n_in,
                              void* d_out, int out_size, void* d_ws,
                              size_t ws_size, hipStream_t stream) {
  (void)in_sizes; (void)n_in; (void)out_size; (void)ws_size;
  const float* h   = (const float*)d_in[0];   // (8,2048,128) f32
  const int*   adj = (const int*)d_in[1];     // (8,2048,2048) i32
  const float* W   = (const float*)d_in[2];   // (128,64) f32
  const float* a   = (const float*)d_in[3];   // (128,1) f32
  float* out = (float*)d_out;                 // (8,2048,64) f32

  // workspace: Wh(4MB) + f1/f2/rmax/rsum (64KB each) = 4.25 MB
  float* Wh   = (float*)d_ws;
  float* f1   = Wh + (size_t)NB * NNODE * OUT_F;
  float* f2   = f1 + NB * NNODE;
  float* rmax = f2 + NB * NNODE;
  float* rsum = rmax + NB * NNODE;

  const int ROWS = NB * NNODE;                          // 16384
  k1_wh   <<<dim3(ROWS / 128), dim3(256), 0, stream>>>(h, W, Wh);
  k2_f    <<<dim3(ROWS / 8),   dim3(256), 0, stream>>>(Wh, a, f1, f2);
  k3_stats<<<dim3(ROWS / 8),   dim3(256), 0, stream>>>(adj, f1, f2, rmax, rsum);
  k4_out  <<<dim3(ROWS / 128), dim3(256), 0, stream>>>(adj, Wh, f1, f2, rmax,
                                                       rsum, out);
}